// Augment_21586505630171
// MI455X (gfx1250) — compile-verified
//
#include <hip/hip_runtime.h>
#include <math.h>

// ---------------- constants ----------------
#define B_     16
#define NS_    320000
#define T_     1001
#define TP_    1008          // T padded to multiple of 16
#define F_     641
#define FP_    656           // F padded to multiple of 16
#define NFFT_  1280
#define HOP_   320
#define PAD_   640
#define SRATE_ 16000.0f
#define NCODE_ 32
#define TWO_PI_ 6.28318530717958647692f

typedef __attribute__((ext_vector_type(2))) float v2f;
typedef __attribute__((ext_vector_type(8))) float v8f;

__device__ __forceinline__ v8f wmma4(v2f a, v2f b, v8f c) {
  // D = A(16x4) * B(4x16) + C(16x16), f32 WMMA
  return __builtin_amdgcn_wmma_f32_16x16x4_f32(false, a, false, b, (short)0, c, false, false);
}

__device__ __forceinline__ size_t sidx(int b, int k, int t) {
  return ((size_t)b * FP_ + (size_t)k) * TP_ + (size_t)t;
}
__device__ __forceinline__ float hannw(int o) {
  return 0.5f - 0.5f * __cosf(TWO_PI_ * (float)o / (float)NFFT_);
}
__device__ __forceinline__ int imin(int a, int b) { return a < b ? a : b; }

// ============================================================
// Per-wave 1280-point complex FFT, LDS resident.
// 1280 = 16 x 80, 80 = 16 x 5.
//   Stage 1: Y[k1][n2]   = sum_n1 D16[k1][n1] x[80*n1+n2]        (WMMA)
//            twiddle W_1280^{n2*k1}
//   Stage 2: U_r[k2a][j] = sum_m D16[k2a][m] Y[j][5m+r]          (WMMA)
//            twiddle W_80^{r*k2a}
//   Stage 3: X[256*k2b + 16*k2a + k1] = sum_r W_5^{r*k2b} U_r    (VALU)
//
// All LDS arrays are declared *inside* this routine so the compiler
// sees addrspace(3) directly (no generic-pointer null checks on ds ops).
// loadfn(o, re, im): produce input sample o.
// storefn(k, re, im): consume output bin k for k in [0, outN).
// ============================================================
template <class LoadFn, class StoreFn>
__device__ __forceinline__ void frame_fft1280(int lane, LoadFn loadfn,
                                              int outN, StoreFn storefn)
{
  __shared__ float xr[NFFT_], xi[NFFT_], sr[NFFT_], si[NFFT_];
  __shared__ float dc[256], dsn[256];
  const int hi = lane >> 4;    // 0 or 1
  const int lm = lane & 15;

  // D16 tables: dc = cos(2 pi k m / 16), dsn = -sin(...)
  for (int i = lane; i < 256; i += 32) {
    float s, c;
    __sincosf(-TWO_PI_ * (float)((i >> 4) * (i & 15)) / 16.0f, &s, &c);
    dc[i] = c; dsn[i] = s;
  }
  for (int o = lane; o < NFFT_; o += 32) {
    float re, im;
    loadfn(o, re, im);
    xr[o] = re; xi[o] = im;
  }
  __syncthreads();

  // Hoisted A fragments (depend only on K-chunk)
  v2f arv[4], aiv[4];
  #pragma unroll
  for (int q = 0; q < 4; ++q) {
    const int k0 = q * 4 + hi * 2;
    arv[q].x = dc[lm * 16 + k0];   arv[q].y = dc[lm * 16 + k0 + 1];
    aiv[q].x = dsn[lm * 16 + k0];  aiv[q].y = dsn[lm * 16 + k0 + 1];
  }

  // -------- Stage 1: 5 column tiles of 16 --------
  for (int c = 0; c < 5; ++c) {
    v8f cr = {0.f,0.f,0.f,0.f,0.f,0.f,0.f,0.f};
    v8f ci = {0.f,0.f,0.f,0.f,0.f,0.f,0.f,0.f};
    #pragma unroll
    for (int q = 0; q < 4; ++q) {
      const int k0 = q * 4 + hi * 2;
      v2f br, bi;
      br.x = xr[k0 * 80 + c * 16 + lm];
      br.y = xr[(k0 + 1) * 80 + c * 16 + lm];
      bi.x = xi[k0 * 80 + c * 16 + lm];
      bi.y = xi[(k0 + 1) * 80 + c * 16 + lm];
      const v2f nbi = -bi;
      cr = wmma4(arv[q], br, cr);
      cr = wmma4(aiv[q], nbi, cr);
      ci = wmma4(arv[q], bi, ci);
      ci = wmma4(aiv[q], br, ci);
    }
    #pragma unroll
    for (int v = 0; v < 8; ++v) {
      const int k1 = v + hi * 8;
      const int n2 = c * 16 + lm;
      float ts, tc;
      __sincosf(-TWO_PI_ * (float)(k1 * n2) * (1.0f / 1280.0f), &ts, &tc);
      const float yr = cr[v], yi = ci[v];
      sr[k1 * 80 + n2] = yr * tc - yi * ts;
      si[k1 * 80 + n2] = yr * ts + yi * tc;
    }
  }
  __syncthreads();

  // -------- Stage 2: 5 sub-sequences r --------
  for (int r = 0; r < 5; ++r) {
    v8f cr = {0.f,0.f,0.f,0.f,0.f,0.f,0.f,0.f};
    v8f ci = {0.f,0.f,0.f,0.f,0.f,0.f,0.f,0.f};
    #pragma unroll
    for (int q = 0; q < 4; ++q) {
      const int k0 = q * 4 + hi * 2;
      v2f br, bi;
      br.x = sr[lm * 80 + 5 * k0 + r];
      br.y = sr[lm * 80 + 5 * (k0 + 1) + r];
      bi.x = si[lm * 80 + 5 * k0 + r];
      bi.y = si[lm * 80 + 5 * (k0 + 1) + r];
      const v2f nbi = -bi;
      cr = wmma4(arv[q], br, cr);
      cr = wmma4(aiv[q], nbi, cr);
      ci = wmma4(arv[q], bi, ci);
      ci = wmma4(aiv[q], br, ci);
    }
    #pragma unroll
    for (int v = 0; v < 8; ++v) {
      const int k2a = v + hi * 8;
      const int j   = lm;  // = k1
      float ts, tc;
      __sincosf(-TWO_PI_ * (float)(r * k2a) * (1.0f / 80.0f), &ts, &tc);
      const float ur = cr[v], ui = ci[v];
      xr[(r * 16 + k2a) * 16 + j] = ur * tc - ui * tc * 0.f - ui * ts; // ur*tc - ui*ts
      xr[(r * 16 + k2a) * 16 + j] = ur * tc - ui * ts;
      xi[(r * 16 + k2a) * 16 + j] = ur * ts + ui * tc;
    }
  }
  __syncthreads();

  // -------- Stage 3: radix-5, constant twiddles W_5^{(r*k2b) mod 5} --------
  // cos(-2 pi m / 5), sin(-2 pi m / 5) for m = 0..4
  const float C5[5] = { 1.0f,  0.30901699437f, -0.80901699437f,
                        -0.80901699437f, 0.30901699437f };
  const float S5[5] = { 0.0f, -0.95105651629f, -0.58778525229f,
                         0.58778525229f, 0.95105651629f };
  for (int p = lane; p < 256; p += 32) {
    const int k2a = p >> 4;
    const int k1  = p & 15;
    float ur[5], ui[5];
    #pragma unroll
    for (int r = 0; r < 5; ++r) {
      ur[r] = xr[(r * 16 + k2a) * 16 + k1];
      ui[r] = xi[(r * 16 + k2a) * 16 + k1];
    }
    #pragma unroll
    for (int k2b = 0; k2b < 5; ++k2b) {
      float accr = 0.f, acci = 0.f;
      #pragma unroll
      for (int r = 0; r < 5; ++r) {
        const int m = (r * k2b) % 5;
        accr += ur[r] * C5[m] - ui[r] * S5[m];
        acci += ur[r] * S5[m] + ui[r] * C5[m];
      }
      const int k = 256 * k2b + 16 * k2a + k1;
      sr[k] = accr;
      si[k] = acci;
    }
  }
  __syncthreads();

  for (int k = lane; k < outN; k += 32) storefn(k, sr[k], si[k]);
}

// ============================================================
// EQ filter frequency responses  ->  eqR/eqI [B][641]
// ============================================================
__device__ __forceinline__ void cmul_acc(float& hr, float& hi,
                                         float nr, float ni, float dr, float di)
{
  // H *= (nr+i ni)/(dr+i di)
  const float inv = 1.0f / (dr * dr + di * di);
  const float qr = (nr * dr + ni * di) * inv;
  const float qi = (ni * dr - nr * di) * inv;
  const float tr = hr * qr - hi * qi;
  const float ti = hr * qi + hi * qr;
  hr = tr; hi = ti;
}

__global__ void eq_kernel(const float* __restrict__ power,
                          const float* __restrict__ gain_u,
                          float* __restrict__ eqR, float* __restrict__ eqI)
{
  const int id = blockIdx.x * blockDim.x + threadIdx.x;
  if (id >= B_ * F_) return;
  const int b = id / F_, k = id % F_;

  const float th = TWO_PI_ * (float)k / (float)NFFT_;
  const float zr = cosf(th),  zi = -sinf(th);               // z = e^{-i th}
  const float z2r = zr * zr - zi * zi, z2i = 2.f * zr * zi; // z^2

  float Hr = 1.f, Hi = 0.f;

  // 8 peaking EQs
  #pragma unroll
  for (int j = 0; j < 8; ++j) {
    const float q = 2.0f * powf(2.5f, power[b * 10 + j]);
    const float g = gain_u[b * 8 + j] * 24.0f - 12.0f;
    const float A = powf(10.0f, g / 40.0f);
    const float center = 60.0f * powf(10000.0f / 60.0f, (float)j / 7.0f);
    const float om = TWO_PI_ * center / SRATE_;
    const float co = cosf(om), si = sinf(om);
    const float al = si / (2.0f * q);
    const float a0 = 1.f + al / A, a1 = -2.f * co, a2 = 1.f - al / A;
    const float b0 = 1.f + al * A, b1 = -2.f * co, b2 = 1.f - al * A;
    const float nr = b0 + b1 * zr + b2 * z2r, ni = b1 * zi + b2 * z2i;
    const float dr = a0 + a1 * zr + a2 * z2r, di = a1 * zi + a2 * z2i;
    cmul_acc(Hr, Hi, nr, ni, dr, di);
  }
  // low shelf @60
  {
    const float q = 2.0f * powf(2.5f, power[b * 10 + 8]);
    const float om = TWO_PI_ * 60.0f / SRATE_;
    const float co = cosf(om), si = sinf(om);
    const float al = si / (2.0f * q);
    const float a0 = 1.f + al, a1 = -2.f * co, a2 = 1.f - al;
    const float b0 = (1.f - co) * 0.5f, b1 = (1.f - co), b2 = (1.f - co) * 0.5f;
    const float nr = b0 + b1 * zr + b2 * z2r, ni = b1 * zi + b2 * z2i;
    const float dr = a0 + a1 * zr + a2 * z2r, di = a1 * zi + a2 * z2i;
    cmul_acc(Hr, Hi, nr, ni, dr, di);
  }
  // high shelf @10000
  {
    const float q = 2.0f * powf(2.5f, power[b * 10 + 9]);
    const float om = TWO_PI_ * 10000.0f / SRATE_;
    const float co = cosf(om), si = sinf(om);
    const float al = si / (2.0f * q);
    const float a0 = 1.f + al, a1 = -2.f * co, a2 = 1.f - al;
    const float b0 = (1.f + co) * 0.5f, b1 = -(1.f + co), b2 = (1.f + co) * 0.5f;
    const float nr = b0 + b1 * zr + b2 * z2r, ni = b1 * zi + b2 * z2i;
    const float dr = a0 + a1 * zr + a2 * z2r, di = a1 * zi + a2 * z2i;
    cmul_acc(Hr, Hi, nr, ni, dr, di);
  }
  eqR[id] = Hr;
  eqI[id] = Hi;
}

// ============================================================
// STFT: one wave per frame; reflect pad + Hann; WMMA FFT;
// fused EQ filter apply.  spec layout [B][FP][TP].
// ============================================================
__global__ void __launch_bounds__(32) stft_kernel(
    const float* __restrict__ wavs,
    const float* __restrict__ eqR, const float* __restrict__ eqI,
    float* __restrict__ specR, float* __restrict__ specI)
{
  const int bt = blockIdx.x;
  const int b = bt / T_, t = bt % T_;
  const int lane = threadIdx.x;
  const float* wv = wavs + (size_t)b * NS_;

  frame_fft1280(lane,
    [&](int o, float& re, float& im) {
      int q = t * HOP_ + o - PAD_;
      q = (q < 0) ? -q : q;
      q = (q >= NS_) ? (2 * NS_ - 2 - q) : q;
      __builtin_prefetch(&wv[q], 0, 0);
      re = wv[q] * hannw(o);
      im = 0.f;
    },
    F_,
    [&](int k, float re, float im) {
      const float er = eqR[b * F_ + k], ei = eqI[b * F_ + k];
      specR[sidx(b, k, t)] = re * er - im * ei;
      specI[sidx(b, k, t)] = re * ei + im * er;
    });
}

// ============================================================
// norm[b][t] = max(mean_k |spec|, 1e-7)
// ============================================================
__global__ void __launch_bounds__(32) norm_kernel(
    const float* __restrict__ specR, const float* __restrict__ specI,
    float* __restrict__ normBuf)
{
  const int bt = blockIdx.x;
  const int b = bt / T_, t = bt % T_;
  const int lane = threadIdx.x;
  float sum = 0.f;
  for (int k = lane; k < F_; k += 32) {
    const float r = specR[sidx(b, k, t)], i = specI[sidx(b, k, t)];
    sum += sqrtf(r * r + i * i);
  }
  for (int off = 16; off > 0; off >>= 1) sum += __shfl_down(sum, off, 32);
  if (lane == 0) normBuf[b * TP_ + t] = fmaxf(sum / (float)F_, 1e-7f);
}

// ============================================================
// Autocorrelation lags 0..32 as cosine-matrix matmul (WMMA):
// corr[b][t][n] = sum_k w_k cos(2 pi k n / 1280) * |spec/norm|^2
// grid: (TP/16, 3 lag tiles, B); one wave per 16x16 tile
// ============================================================
__global__ void __launch_bounds__(32) corr_kernel(
    const float* __restrict__ specR, const float* __restrict__ specI,
    const float* __restrict__ normBuf, float* __restrict__ corr)
{
  const int b = blockIdx.z;
  const int mt = blockIdx.y;            // lag tile (rows 16*mt .. +15)
  const int tt = blockIdx.x;            // time tile
  const int lane = threadIdx.x;
  const int hi = lane >> 4, lm = lane & 15;
  const int tcol = tt * 16 + lm;
  const int lagA = mt * 16 + lm;        // A-matrix row this lane supplies

  float invn2 = 0.f;
  if (tcol < T_) {
    const float nv = normBuf[b * TP_ + tcol];
    invn2 = 1.0f / (nv * nv);
  }

  const float theta = TWO_PI_ / (float)NFFT_;
  v8f acc = {0.f,0.f,0.f,0.f,0.f,0.f,0.f,0.f};

  for (int kk = 0; kk < 644; kk += 4) {
    const int k0 = kk + hi * 2;
    v2f a, bf;
    float w0 = 0.f, w1 = 0.f;
    if (k0 < F_)     w0 = ((k0 == 0 || k0 == 640) ? 1.0f : 2.0f) * (1.0f / 1280.0f);
    if (k0 + 1 < F_) w1 = ((k0 + 1 == 640) ? 1.0f : 2.0f) * (1.0f / 1280.0f);
    a.x = w0 * __cosf(theta * (float)(k0 * lagA));
    a.y = w1 * __cosf(theta * (float)((k0 + 1) * lagA));

    float p0 = 0.f, p1 = 0.f;
    if (tcol < T_) {
      if (k0 < F_) {
        const float r = specR[sidx(b, k0, tcol)], i = specI[sidx(b, k0, tcol)];
        p0 = (r * r + i * i) * invn2;
      }
      if (k0 + 1 < F_) {
        const float r = specR[sidx(b, k0 + 1, tcol)], i = specI[sidx(b, k0 + 1, tcol)];
        p1 = (r * r + i * i) * invn2;
      }
    }
    bf.x = p0; bf.y = p1;
    acc = wmma4(a, bf, acc);
  }

  #pragma unroll
  for (int v = 0; v < 8; ++v) {
    const int lag = mt * 16 + v + hi * 8;
    if (lag < NCODE_ + 1 && tcol < T_)
      corr[((size_t)b * TP_ + tcol) * 33 + lag] = acc[v];
  }
}

// ============================================================
// Levinson-Durbin, one thread per (b,t)
// ============================================================
__global__ void levinson_kernel(const float* __restrict__ corr,
                                float* __restrict__ lpc)
{
  const int id = blockIdx.x * blockDim.x + threadIdx.x;
  if (id >= B_ * T_) return;
  const int b = id / T_, t = id % T_;
  const float* c = corr + ((size_t)b * TP_ + t) * 33;

  float r[33], sol[33], ns[33];
  for (int i = 0; i < 33; ++i) r[i] = c[i];
  sol[0] = 1.0f;
  sol[1] = -r[1] / fmaxf(r[0], 1e-7f);
  for (int i = 2; i < 33; ++i) sol[i] = 0.f;
  float extra = r[0] + r[1] * sol[1];

  for (int k = 1; k < NCODE_; ++k) {
    float lam = 0.f;
    for (int j = 0; j <= k; ++j) lam += sol[j] * r[k + 1 - j];
    lam = -lam / fmaxf(extra, 1e-7f);
    for (int j = 0; j <= k + 1; ++j) {
      const float a0 = (j <= k) ? sol[j] : 0.f;
      const int jf = k + 1 - j;
      const float a1 = (jf <= k) ? sol[jf] : 0.f;
      ns[j] = a0 + lam * a1;
    }
    for (int j = 0; j <= k + 1; ++j) sol[j] = ns[j];
    extra = (1.0f - lam * lam) * extra;
  }
  float* lp = lpc + ((size_t)b * TP_ + t) * NCODE_;
  for (int n = 0; n < NCODE_; ++n) lp[n] = sol[n + 1];
}

// ============================================================
// LPC envelope via cos/sin matmul (WMMA):
// H(k,t) = 1 + sum_n lpc_n cos(th k (n+1))  - i sum_n lpc_n sin(...)
// filt = 1/|H| (|H|<1e-7 -> 1)
// grid: (TP/16, FP/16, B)
// ============================================================
__global__ void __launch_bounds__(32) env_kernel(
    const float* __restrict__ lpc, float* __restrict__ filt)
{
  const int b = blockIdx.z;
  const int ft = blockIdx.y;
  const int tt = blockIdx.x;
  const int lane = threadIdx.x;
  const int hi = lane >> 4, lm = lane & 15;
  const int kfA = ft * 16 + lm;         // A-matrix freq row for this lane
  const int tcol = tt * 16 + lm;

  const float theta = TWO_PI_ / (float)NFFT_;
  v8f hr = {0.f,0.f,0.f,0.f,0.f,0.f,0.f,0.f};
  v8f hm = {0.f,0.f,0.f,0.f,0.f,0.f,0.f,0.f};

  const float* lp = (tcol < T_) ? (lpc + ((size_t)b * TP_ + tcol) * NCODE_) : nullptr;

  #pragma unroll
  for (int kk = 0; kk < NCODE_; kk += 4) {
    const int k0 = kk + hi * 2;
    v2f ac, as, bf;
    float s0, c0, s1, c1;
    __sincosf(theta * (float)kfA * (float)(k0 + 1), &s0, &c0);
    __sincosf(theta * (float)kfA * (float)(k0 + 2), &s1, &c1);
    ac.x = c0; ac.y = c1;
    as.x = s0; as.y = s1;
    bf.x = lp ? lp[k0] : 0.f;
    bf.y = lp ? lp[k0 + 1] : 0.f;
    hr = wmma4(ac, bf, hr);
    hm = wmma4(as, bf, hm);
  }

  #pragma unroll
  for (int v = 0; v < 8; ++v) {
    const int kf = ft * 16 + v + hi * 8;
    if (kf < F_ && tcol < T_) {
      const float re = 1.0f + hr[v];
      const float im = hm[v];
      float d = sqrtf(re * re + im * im);
      d = (d - 1e-7f < 0.f) ? 1.0f : d;
      filt[sidx(b, kf, tcol)] = 1.0f / d;
    }
  }
}

// ============================================================
// Source/filter split + frequency interpolation + recombine.
// One wave per (b,t) column; column cached in LDS.
// Overwrites specR/specI with the final shifted spectrum.
// ============================================================
__global__ void __launch_bounds__(32) shift_kernel(
    float* __restrict__ specR, float* __restrict__ specI,
    const float* __restrict__ filt,
    const float* __restrict__ shift_u, const int* __restrict__ flip)
{
  __shared__ float fc[F_], mg[F_], an[F_];
  const int bt = blockIdx.x;
  const int b = bt / T_, t = bt % T_;
  const int lane = threadIdx.x;

  float fs = shift_u[b * 2 + 0] * 0.4f + 1.0f;
  float ps = shift_u[b * 2 + 1] * 1.0f + 1.0f;
  if (flip[b * 2 + 0] == 1) fs = 1.0f / fs;
  if (flip[b * 2 + 1] == 1) ps = 1.0f / ps;

  for (int k = lane; k < F_; k += 32) {
    const float f = filt[sidx(b, k, t)];
    const float inv = 1.0f / (f + 1e-7f);
    const float r = specR[sidx(b, k, t)] * inv;
    const float i = specI[sidx(b, k, t)] * inv;
    fc[k] = f;
    mg[k] = sqrtf(r * r + i * i);
    an[k] = atan2f(i, r);
  }
  __syncthreads();

  const float outF = floorf((float)F_ * fs);
  const float outP = floorf((float)F_ * ps);

  for (int i = lane; i < F_; i += 32) {
    // filter interpolation
    float sp = fminf(fmaxf(((float)i + 0.5f) / fs - 0.5f, 0.0f), (float)(F_ - 1));
    int lo = (int)floorf(sp);
    int hi2 = imin(lo + 1, F_ - 1);
    float w = sp - (float)lo;
    float fv = ((float)i < outF) ? (fc[lo] * (1.f - w) + fc[hi2] * w) : 0.f;
    // source (mag/phase) interpolation
    float sp2 = fminf(fmaxf(((float)i + 0.5f) / ps - 0.5f, 0.0f), (float)(F_ - 1));
    int lo2 = (int)floorf(sp2);
    int hi3 = imin(lo2 + 1, F_ - 1);
    float w2 = sp2 - (float)lo2;
    float m = mg[lo2] * (1.f - w2) + mg[hi3] * w2;
    float a = an[lo2] * (1.f - w2) + an[hi3] * w2;
    if (!((float)i < outP)) { m = 0.f; a = 0.f; }
    float sn, cs;
    __sincosf(a, &sn, &cs);
    specR[sidx(b, i, t)] = m * cs * fv;
    specI[sidx(b, i, t)] = m * sn * fv;
  }
}

// ============================================================
// ISTFT: inverse FFT per frame via conj(FFT(conj(S)))/N, windowed.
// frames[b*T+t][0..1279]
// ============================================================
__global__ void __launch_bounds__(32) istft_kernel(
    const float* __restrict__ specR, const float* __restrict__ specI,
    float* __restrict__ frames)
{
  const int bt = blockIdx.x;
  const int b = bt / T_, t = bt % T_;
  const int lane = threadIdx.x;
  float* fr = frames + (size_t)bt * NFFT_;

  frame_fft1280(lane,
    [&](int o, float& re, float& im) {
      const int kk = (o <= 640) ? o : (NFFT_ - o);
      const float r  = specR[sidx(b, kk, t)];
      const float ih = specI[sidx(b, kk, t)];
      const float imS = (o <= 640) ? ih : -ih;  // hermitian extension
      re = r;
      im = -imS;                                 // conj for inverse trick
    },
    NFFT_,
    [&](int k, float re, float im) {
      (void)im;
      fr[k] = re * (1.0f / (float)NFFT_) * hannw(k);
    });
}

// ============================================================
// Deterministic OLA gather + window-square normalization
// ============================================================
__global__ void ola_kernel(const float* __restrict__ frames,
                           float* __restrict__ ola)
{
  const int id = blockIdx.x * blockDim.x + threadIdx.x;
  if (id >= B_ * NS_) return;
  const int b = id / NS_, p = id % NS_;
  const int pos = p + PAD_;

  float acc = 0.f, wsq = 0.f;
  const int tc = pos / HOP_;
  #pragma unroll
  for (int dt = 0; dt < 4; ++dt) {
    const int t = tc - dt;
    if (t < 0 || t >= T_) continue;
    const int o = pos - t * HOP_;
    if (o >= NFFT_) continue;
    acc += frames[((size_t)(b * T_ + t)) * NFFT_ + o];
    const float w = hannw(o);
    wsq += w * w;
  }
  ola[id] = (wsq > 1e-11f) ? (acc / wsq) : acc;
}

// ============================================================
// Per-batch max reduction (2 stage, deterministic) + final scale
// ============================================================
__global__ void max1_kernel(const float* __restrict__ ola,
                            float* __restrict__ part)
{
  __shared__ float sm[256];
  const int b = blockIdx.y, bx = blockIdx.x, tid = threadIdx.x;
  float m = -3.4e38f;
  for (int i = bx * 256 + tid; i < NS_; i += 64 * 256)
    m = fmaxf(m, ola[(size_t)b * NS_ + i]);
  sm[tid] = m;
  __syncthreads();
  for (int s = 128; s > 0; s >>= 1) {
    if (tid < s) sm[tid] = fmaxf(sm[tid], sm[tid + s]);
    __syncthreads();
  }
  if (tid == 0) part[b * 64 + bx] = sm[0];
}

__global__ void max2_kernel(const float* __restrict__ part,
                            float* __restrict__ maxv)
{
  __shared__ float sm[64];
  const int b = blockIdx.x, tid = threadIdx.x;
  sm[tid] = part[b * 64 + tid];
  __syncthreads();
  for (int s = 32; s > 0; s >>= 1) {
    if (tid < s) sm[tid] = fmaxf(sm[tid], sm[tid + s]);
    __syncthreads();
  }
  if (tid == 0) maxv[b] = fmaxf(sm[0], 1e-7f);
}

__global__ void scale_kernel(const float* __restrict__ ola,
                             const float* __restrict__ maxv,
                             float* __restrict__ out)
{
  const int id = blockIdx.x * blockDim.x + threadIdx.x;
  if (id >= B_ * NS_) return;
  const int b = id / NS_;
  out[id] = ola[id] / maxv[b];
}

// ============================================================
// Host-side orchestration
// ============================================================
extern "C" void kernel_launch(void* const* d_in, const int* in_sizes, int n_in,
                              void* d_out, int out_size, void* d_ws, size_t ws_size,
                              hipStream_t stream)
{
  (void)in_sizes; (void)n_in; (void)out_size; (void)ws_size;
  const float* wavs    = (const float*)d_in[0];
  const float* power   = (const float*)d_in[1];
  const float* gain_u  = (const float*)d_in[2];
  const float* shift_u = (const float*)d_in[3];
  const int*   flip    = (const int*)d_in[4];
  float* out = (float*)d_out;

  // workspace carve-up (floats)
  float* ws = (float*)d_ws;
  const size_t SZ_SPEC = (size_t)B_ * FP_ * TP_;     // 10,579,968
  float* specR  = ws;                ws += SZ_SPEC;
  float* specI  = ws;                ws += SZ_SPEC;
  float* filt   = ws;                ws += SZ_SPEC;
  float* corr   = ws;                ws += (size_t)B_ * TP_ * 33;
  float* lpc    = ws;                ws += (size_t)B_ * TP_ * NCODE_;
  float* eqR    = ws;                ws += (size_t)B_ * F_;
  float* eqI    = ws;                ws += (size_t)B_ * F_;
  float* normB  = ws;                ws += (size_t)B_ * TP_;
  float* frames = ws;                ws += (size_t)B_ * T_ * NFFT_;
  float* ola    = ws;                ws += (size_t)B_ * NS_;
  float* part   = ws;                ws += (size_t)B_ * 64;
  float* maxv   = ws;                ws += (size_t)B_;

  // 1) EQ filter responses
  eq_kernel<<<(B_ * F_ + 255) / 256, 256, 0, stream>>>(power, gain_u, eqR, eqI);

  // 2) STFT (WMMA FFT) fused with EQ apply
  stft_kernel<<<B_ * T_, 32, 0, stream>>>(wavs, eqR, eqI, specR, specI);

  // 3) per-column spectral norm
  norm_kernel<<<B_ * T_, 32, 0, stream>>>(specR, specI, normB);

  // 4) autocorrelation lags (WMMA cosine transform)
  corr_kernel<<<dim3(TP_ / 16, 3, B_), 32, 0, stream>>>(specR, specI, normB, corr);

  // 5) Levinson-Durbin LPC
  levinson_kernel<<<(B_ * T_ + 127) / 128, 128, 0, stream>>>(corr, lpc);

  // 6) LPC envelope (WMMA cos/sin matmul)
  env_kernel<<<dim3(TP_ / 16, FP_ / 16, B_), 32, 0, stream>>>(lpc, filt);

  // 7) source/filter split + frequency interpolation + recombine
  shift_kernel<<<B_ * T_, 32, 0, stream>>>(specR, specI, filt, shift_u, flip);

  // 8) ISTFT (WMMA inverse FFT) -> windowed frames
  istft_kernel<<<B_ * T_, 32, 0, stream>>>(specR, specI, frames);

  // 9) deterministic overlap-add with window-square compensation
  ola_kernel<<<(B_ * NS_ + 255) / 256, 256, 0, stream>>>(frames, ola);

  // 10) per-batch max + final normalization
  max1_kernel<<<dim3(64, B_), 256, 0, stream>>>(ola, part);
  max2_kernel<<<B_, 64, 0, stream>>>(part, maxv);
  scale_kernel<<<(B_ * NS_ + 255) / 256, 256, 0, stream>>>(ola, maxv, out);
}